// Conv1D_601295421784
// MI455X (gfx1250) — compile-verified
//
#include <hip/hip_runtime.h>
#include <stdint.h>

// Depthwise causal conv1d: B=4, C=4096, L=4096, K=4, pad=3 both sides.
// out[b,c,t] = bias[c] + sum_{k=0..3} kernel[k,0,c] * x[b,c, t-3+k],  t in [0,4099)
//
// Memory-bound (AI ~ 1 flop/byte): ~525 MB of traffic => ~22.5us floor @ 23.3 TB/s.
// Strategy: stream each row tile through LDS using gfx1250 async global->LDS
// b128 copies (ASYNCcnt path) with NT load hints, then 4 LDS reads + 4 FMAs per
// output, coalesced NT b32 stores. Weights/bias are block-uniform -> scalar loads.

#define B_DIM 4
#define C_DIM 4096
#define L_DIM 4096
#define K_DIM 4
#define OUT_L 4099            // L + 2*3 - 3
#define TILE  1024            // outputs per block (main region)
#define NTHREADS 256

__global__ __launch_bounds__(NTHREADS)
void dwconv1d_async_kernel(const float* __restrict__ x,      // (B,C,L)
                           const float* __restrict__ w,      // (K,1,C) -> w[k*C + c]
                           const float* __restrict__ bias,   // (C)
                           float* __restrict__ out)          // (B,C,OUT_L)
{
    // LDS layout (floats):
    //   [0]            unused (alignment pad)
    //   [1..3]         head halo: x[t0-3 .. t0-1]   (zeros when t0 == 0)
    //   [4..4+TILE-1]  main:      x[t0 .. t0+TILE-1]  (async b128, 16B aligned at lds[4])
    //   [4+TILE .. 4+TILE+2] tail zeros (used only by the last tile's padded outputs)
    __shared__ __align__(16) float lds[TILE + 8];

    const int row = blockIdx.y;            // row = b*C + c
    const int c   = row & (C_DIM - 1);
    const int t0  = blockIdx.x * TILE;     // tile start along output/time axis
    const int tid = threadIdx.x;

    const float* xrow = x + (size_t)row * L_DIM;

    // ---- async copy of main tile: x[t0 .. t0+1023] -> lds[4..1027]
    // One b128 (16B) transfer per lane; 256 lanes * 16B = 4096B = TILE floats.
    // GVS addressing: 64-bit SGPR base + 32-bit per-lane byte offset.
    // NT hint: each cacheline of x is consumed exactly once -> evict-first.
    {
        uint32_t ldsOff = ((uint32_t)(uintptr_t)&lds[4]) + (uint32_t)tid * 16u;
        uint32_t gOff   = (uint32_t)(t0 * 4) + (uint32_t)tid * 16u;
        asm volatile("global_load_async_to_lds_b128 %0, %1, %2 th:TH_LOAD_NT"
                     :
                     : "v"(ldsOff), "v"(gOff), "s"(xrow)
                     : "memory");
    }

    // ---- head halo (guarded) + tail zeros
    if (tid < 3) {
        int gx = t0 - 3 + tid;
        lds[1 + tid] = (gx >= 0) ? xrow[gx] : 0.0f;   // head: x[t0-3..t0-1]
        lds[4 + TILE + tid] = 0.0f;                   // tail zeros for padded outputs
    }

    // Per-channel taps + bias: uniform across the block -> scalar loads.
    const float w0 = w[0 * C_DIM + c];
    const float w1 = w[1 * C_DIM + c];
    const float w2 = w[2 * C_DIM + c];
    const float w3 = w[3 * C_DIM + c];
    const float bc = bias[c];

    // Wave-level: async transfers done; then workgroup barrier for cross-wave LDS visibility.
    asm volatile("s_wait_asynccnt 0x0" ::: "memory");
    __syncthreads();

    // Last tile also produces the 3 padded-tail outputs (t = 4096..4098).
    const int nOut = (t0 + TILE >= L_DIM) ? (OUT_L - t0) : TILE;
    float* __restrict__ orow = out + (size_t)row * OUT_L + t0;

    // x[t0 + j] == lds[4 + j]  =>  y[t0+i] = sum_k w_k * lds[i + 1 + k]
    for (int i = tid; i < nOut; i += NTHREADS) {
        float acc = bc;
        acc = fmaf(w0, lds[i + 1], acc);
        acc = fmaf(w1, lds[i + 2], acc);
        acc = fmaf(w2, lds[i + 3], acc);
        acc = fmaf(w3, lds[i + 4], acc);
        __builtin_nontemporal_store(acc, &orow[i]);   // coalesced NT b32 store
    }
}

extern "C" void kernel_launch(void* const* d_in, const int* in_sizes, int n_in,
                              void* d_out, int out_size, void* d_ws, size_t ws_size,
                              hipStream_t stream) {
    (void)in_sizes; (void)n_in; (void)d_ws; (void)ws_size; (void)out_size;
    const float* x    = (const float*)d_in[0];   // (B,C,L) fp32
    const float* w    = (const float*)d_in[1];   // (K,1,C) fp32
    const float* bias = (const float*)d_in[2];   // (C)     fp32
    float* out        = (float*)d_out;           // (B,C,OUT_L) fp32

    dim3 grid(L_DIM / TILE, B_DIM * C_DIM);      // 4 tiles x 16384 rows
    dwconv1d_async_kernel<<<grid, NTHREADS, 0, stream>>>(x, w, bias, out);
}